// HierarchicalMamba_83451214561954
// MI455X (gfx1250) — compile-verified
//
#include <hip/hip_runtime.h>
#include <hip/hip_bf16.h>
#include <math.h>

// ---------------------------------------------------------------------------
// HierarchicalMamba for MI455X (gfx1250, wave32).
// fp32 throughout (workload is bandwidth/latency-bound, not matrix-rate bound)
// All dense projections AND the attention core run on V_WMMA_F32_16X16X4_F32.
// ---------------------------------------------------------------------------

typedef float v2f __attribute__((ext_vector_type(2)));
typedef float v8f __attribute__((ext_vector_type(8)));

#define D_MODEL   128
#define D_INNER   256
#define D_STATE   16
#define DT_RANK   8
#define NPROJ     40   // DT_RANK + 2*D_STATE
#define NPROJ_PAD 48   // padded to a 16-multiple for the WMMA GEMM
#define BATCH     4
#define SEQ       4096
#define N_HEADS   8
#define HEAD_D    16

// ---------------------------------------------------------------------------
// Generic WMMA GEMM:  C[M,N] = A[M,K] (row-major, lda) x W[N,K]^T (row-major,
// ldw) [+ R].  One wave computes one 16x16 C tile, K-step 4.
// f32 16x16x4 fragments (ISA 7.12.2):
//   A: lane<16 row=lane v0=K+0,v1=K+1 ; lane>=16 row=lane-16 v0=K+2,v1=K+3
//   B mirrored (K by half, N across lanes) ; C/D: vgpr r -> M=r+8*half, N=lane&15
// ---------------------------------------------------------------------------
__global__ void gemm16_wmma_f32(const float* __restrict__ A, int lda,
                                const float* __restrict__ W, int ldw,
                                float* __restrict__ C, int ldc,
                                const float* __restrict__ R,
                                int M, int N, int K) {
    const int lane = threadIdx.x;            // 0..31
    const int wave = threadIdx.y;
    const int n0 = blockIdx.x * 16;
    const int m0 = (blockIdx.y * blockDim.y + wave) * 16;
    if (m0 >= M || n0 >= N) return;          // wave-uniform exit (EXEC stays full)

    const int row  = lane & 15;
    const int half = lane >> 4;              // selects K pair
    const float* __restrict__ Ap = A + (size_t)(m0 + row) * lda + 2 * half;
    const float* __restrict__ Wp = W + (size_t)(n0 + row) * ldw + 2 * half;

    v8f acc = {0.f, 0.f, 0.f, 0.f, 0.f, 0.f, 0.f, 0.f};
    for (int k = 0; k < K; k += 4) {
        v2f a, b;
        a.x = Ap[k];  a.y = Ap[k + 1];
        b.x = Wp[k];  b.y = Wp[k + 1];
        acc = __builtin_amdgcn_wmma_f32_16x16x4_f32(
            false, a, false, b, (short)0, acc, false, false);
    }
    const int cn = n0 + row;
#pragma unroll
    for (int r = 0; r < 8; ++r) {
        const int cm = m0 + r + half * 8;
        float v = acc[r];
        if (R) v += R[(size_t)cm * ldc + cn];
        C[(size_t)cm * ldc + cn] = v;
    }
}

// ---------------------------------------------------------------------------
// gctx[b, s, :] = x[b, 8*s, :]
// ---------------------------------------------------------------------------
__global__ void gather8_kernel(const float* __restrict__ x, float* __restrict__ g) {
    int idx = blockIdx.x * blockDim.x + threadIdx.x;
    if (idx >= BATCH * 512 * D_MODEL) return;
    int c = idx & (D_MODEL - 1);
    int s = (idx >> 7) & 511;
    int b = idx >> 16;                        // 512*128 per batch
    g[idx] = x[((size_t)b * SEQ + (size_t)s * 8) * D_MODEL + c];
}

// ---------------------------------------------------------------------------
// wp[48 x 256]: rows <40 copied from xproj_w, rows 40..47 zero.
// ---------------------------------------------------------------------------
__global__ void pad_w_kernel(const float* __restrict__ w, float* __restrict__ wp) {
    int idx = blockIdx.x * blockDim.x + threadIdx.x;
    if (idx >= NPROJ_PAD * D_INNER) return;
    int j = idx >> 8;
    wp[idx] = (j < NPROJ) ? w[idx] : 0.f;     // contiguous copy works: idx = j*256+k
}

// ---------------------------------------------------------------------------
// Depthwise causal conv (width 4) over xm (= xz cols [0,256)) + SiLU.
// ---------------------------------------------------------------------------
__global__ void conv_silu_kernel(const float* __restrict__ xz, int ldxz,
                                 const float* __restrict__ cw,
                                 const float* __restrict__ cb,
                                 float* __restrict__ xc,
                                 int L, int rows) {
    int idx = blockIdx.x * blockDim.x + threadIdx.x;
    if (idx >= rows * D_INNER) return;
    int d = idx & (D_INNER - 1);
    int r = idx >> 8;
    int l = r % L;
    float s = cb[d];
#pragma unroll
    for (int k = 0; k < 4; ++k) {
        int ls = l - 3 + k;
        if (ls >= 0)
            s += xz[(size_t)(r - l + ls) * ldxz + d] * cw[d * 4 + k];
    }
    s = s / (1.f + __expf(-s));               // silu
    xc[(size_t)r * D_INNER + d] = s;
}

// ---------------------------------------------------------------------------
// dt = softplus(dt_lin + dt_b), in place.
// ---------------------------------------------------------------------------
__global__ void softplus_bias_kernel(float* __restrict__ dtb,
                                     const float* __restrict__ bias, int rows) {
    int idx = blockIdx.x * blockDim.x + threadIdx.x;
    if (idx >= rows * D_INNER) return;
    int d = idx & (D_INNER - 1);
    float x = dtb[idx] + bias[d];
    dtb[idx] = (x > 20.f) ? x : log1pf(__expf(x));
}

// ---------------------------------------------------------------------------
// Segmented selective scan, phase 1: per (chunk, segment, d) final segment
// state from h=0 plus sum(dt)  (prod exp(dt_t*A) == exp(A * sum dt_t)).
// ---------------------------------------------------------------------------
__global__ void scan_pass1_kernel(const float* __restrict__ dt,
                                  const float* __restrict__ xc,
                                  const float* __restrict__ dbl,
                                  const float* __restrict__ A_log,
                                  float* __restrict__ hseg,
                                  float* __restrict__ sumdt,
                                  int L, int SL, int nseg) {
    int b   = blockIdx.x / nseg;
    int seg = blockIdx.x % nseg;
    int d   = threadIdx.x;
    float A[D_STATE];
#pragma unroll
    for (int n = 0; n < D_STATE; ++n) A[n] = -__expf(A_log[d * D_STATE + n]);
    float h[D_STATE];
#pragma unroll
    for (int n = 0; n < D_STATE; ++n) h[n] = 0.f;
    float sd = 0.f;
    const int l0 = seg * SL;
    for (int l = l0; l < l0 + SL; ++l) {
        size_t r = (size_t)b * L + l;
        float dtv = dt[r * D_INNER + d];
        float xv  = xc[r * D_INNER + d];
        float du  = dtv * xv;
        sd += dtv;
        const float* Bp = dbl + r * NPROJ_PAD + DT_RANK;
#pragma unroll
        for (int n = 0; n < D_STATE; ++n)
            h[n] = __expf(dtv * A[n]) * h[n] + du * Bp[n];
    }
    size_t o = (size_t)blockIdx.x * D_INNER + d;
#pragma unroll
    for (int n = 0; n < D_STATE; ++n) hseg[o * D_STATE + n] = h[n];
    sumdt[o] = sd;
}

// Phase 2 (tiny): sequential combine across segments -> carry-in per segment.
__global__ void scan_combine_kernel(const float* __restrict__ hseg,
                                    const float* __restrict__ sumdt,
                                    const float* __restrict__ A_log,
                                    float* __restrict__ hin, int nseg) {
    int b = blockIdx.x;
    int d = threadIdx.x;
    float A[D_STATE];
#pragma unroll
    for (int n = 0; n < D_STATE; ++n) A[n] = -__expf(A_log[d * D_STATE + n]);
    float h[D_STATE];
#pragma unroll
    for (int n = 0; n < D_STATE; ++n) h[n] = 0.f;
    for (int s = 0; s < nseg; ++s) {
        size_t o = ((size_t)(b * nseg + s) * D_INNER + d);
#pragma unroll
        for (int n = 0; n < D_STATE; ++n) hin[o * D_STATE + n] = h[n];
        float p_sd = sumdt[o];
#pragma unroll
        for (int n = 0; n < D_STATE; ++n)
            h[n] = __expf(A[n] * p_sd) * h[n] + hseg[o * D_STATE + n];
    }
}

// Phase 3: replay each segment with its carry-in, emit y = (scan + x*D)*silu(z).
__global__ void scan_pass2_kernel(const float* __restrict__ dt,
                                  const float* __restrict__ xc,
                                  const float* __restrict__ dbl,
                                  const float* __restrict__ A_log,
                                  const float* __restrict__ Dp,
                                  const float* __restrict__ xz, int ldxz,
                                  const float* __restrict__ hin,
                                  float* __restrict__ y,
                                  int L, int SL, int nseg) {
    int b   = blockIdx.x / nseg;
    int seg = blockIdx.x % nseg;
    int d   = threadIdx.x;
    float A[D_STATE];
#pragma unroll
    for (int n = 0; n < D_STATE; ++n) A[n] = -__expf(A_log[d * D_STATE + n]);
    float h[D_STATE];
    size_t o = (size_t)blockIdx.x * D_INNER + d;
#pragma unroll
    for (int n = 0; n < D_STATE; ++n) h[n] = hin[o * D_STATE + n];
    float Dv = Dp[d];
    const int l0 = seg * SL;
    for (int l = l0; l < l0 + SL; ++l) {
        size_t r = (size_t)b * L + l;
        float dtv = dt[r * D_INNER + d];
        float xv  = xc[r * D_INNER + d];
        float du  = dtv * xv;
        const float* Bp = dbl + r * NPROJ_PAD + DT_RANK;
        const float* Cp = Bp + D_STATE;
        float yv = 0.f;
#pragma unroll
        for (int n = 0; n < D_STATE; ++n) {
            h[n] = __expf(dtv * A[n]) * h[n] + du * Bp[n];
            yv += h[n] * Cp[n];
        }
        yv += xv * Dv;
        float zv = xz[r * (size_t)ldxz + D_INNER + d];
        yv *= zv / (1.f + __expf(-zv));       // * silu(z)
        y[r * D_INNER + d] = yv;
    }
}

// ---------------------------------------------------------------------------
// WMMA flash attention: one wave per (16-row q-tile, head).  Per key tile:
// 4 WMMA for S = Q K^T, online softmax (row stats live in the 16-lane halves
// of the D fragment; xor-shuffles 1/2/4/8 reduce in-half), P spilled through
// a per-wave LDS tile (stride 18 -> conflict-free, 8B aligned) and reloaded
// in A-fragment order, 4 WMMA for O += P V.
// ---------------------------------------------------------------------------
#define PL_STRIDE 18
__global__ __launch_bounds__(128)
void attn_wmma_kernel(const float* __restrict__ q, const float* __restrict__ k,
                      const float* __restrict__ v, float* __restrict__ o) {
    __shared__ float plds[4][16 * PL_STRIDE];
    const int lane = threadIdx.x;
    const int wave = threadIdx.y;
    const int row  = lane & 15;
    const int half = lane >> 4;
    const int h    = blockIdx.y;
    const int qt   = blockIdx.x * 4 + wave;   // q tile index (0..1023)
    const int rq0  = qt * 16;
    const int b    = qt >> 8;                 // 256 q-tiles per batch
    const float* qp = q + (size_t)(rq0 + row) * D_MODEL + h * HEAD_D;
    const float* Kb = k + (size_t)b * 512 * D_MODEL + h * HEAD_D;
    const float* Vb = v + (size_t)b * 512 * D_MODEL + h * HEAD_D;
    float* pl = plds[wave];

    v2f qa[4];
#pragma unroll
    for (int t = 0; t < 4; ++t) {
        int ka = 4 * t + 2 * half;
        qa[t].x = qp[ka]; qa[t].y = qp[ka + 1];
    }
    v8f oacc = {0.f, 0.f, 0.f, 0.f, 0.f, 0.f, 0.f, 0.f};
    float mrow[8], ssum[8];
#pragma unroll
    for (int r = 0; r < 8; ++r) { mrow[r] = -INFINITY; ssum[r] = 0.f; }

    for (int kt = 0; kt < 32; ++kt) {
        // S tile = Q @ K^T
        v8f sacc = {0.f, 0.f, 0.f, 0.f, 0.f, 0.f, 0.f, 0.f};
#pragma unroll
        for (int t = 0; t < 4; ++t) {
            int ka = 4 * t + 2 * half;
            const float* kp = Kb + (size_t)(kt * 16 + row) * D_MODEL + ka;
            v2f bk; bk.x = kp[0]; bk.y = kp[1];
            sacc = __builtin_amdgcn_wmma_f32_16x16x4_f32(
                false, qa[t], false, bk, (short)0, sacc, false, false);
        }
        // online softmax per row (r + 8*half); keys spread over the 16 lanes
        float corr[8];
#pragma unroll
        for (int r = 0; r < 8; ++r) {
            float s = sacc[r] * 0.25f;        // 1/sqrt(16)
            float mx = s;
#pragma unroll
            for (int msk = 1; msk < 16; msk <<= 1)
                mx = fmaxf(mx, __shfl_xor(mx, msk, 32));
            float nm = fmaxf(mrow[r], mx);
            float p  = __expf(s - nm);
            float ps = p;
#pragma unroll
            for (int msk = 1; msk < 16; msk <<= 1)
                ps += __shfl_xor(ps, msk, 32);
            corr[r] = __expf(mrow[r] - nm);
            ssum[r] = ssum[r] * corr[r] + ps;
            mrow[r] = nm;
            pl[(r + 8 * half) * PL_STRIDE + row] = p;   // D-layout -> LDS
        }
        asm volatile("s_wait_dscnt 0" ::: "memory");    // cross-lane LDS visibility
#pragma unroll
        for (int r = 0; r < 8; ++r) oacc[r] *= corr[r];
        // O += P @ V  (P reloaded from LDS in A-fragment order)
#pragma unroll
        for (int t = 0; t < 4; ++t) {
            int ka = 4 * t + 2 * half;
            v2f pa; pa.x = pl[row * PL_STRIDE + ka];
                    pa.y = pl[row * PL_STRIDE + ka + 1];
            const float* vp = Vb + (size_t)(kt * 16 + ka) * D_MODEL + row;
            v2f bv; bv.x = vp[0]; bv.y = vp[D_MODEL];
            oacc = __builtin_amdgcn_wmma_f32_16x16x4_f32(
                false, pa, false, bv, (short)0, oacc, false, false);
        }
        asm volatile("s_wait_dscnt 0" ::: "memory");    // reads done before reuse
    }
    float* op = o + (size_t)rq0 * D_MODEL + h * HEAD_D;
#pragma unroll
    for (int r = 0; r < 8; ++r)
        op[(size_t)(r + 8 * half) * D_MODEL + row] = oacc[r] / ssum[r];
}

// ---------------------------------------------------------------------------
// Host side
// ---------------------------------------------------------------------------
static inline void launch_gemm(const float* A, int lda, const float* W, int ldw,
                               float* C, int ldc, const float* R,
                               int M, int N, int K, hipStream_t stream) {
    dim3 blk(32, 4);
    dim3 grd(N / 16, (M + 63) / 64);
    gemm16_wmma_f32<<<grd, blk, 0, stream>>>(A, lda, W, ldw, C, ldc, R, M, N, K);
}

static void run_mamba(const float* xin, int rows, int L, int SL,
                      const float* in_w, const float* conv_w, const float* conv_b,
                      const float* xproj_w, const float* dt_w, const float* dt_b,
                      const float* A_log, const float* Dp, const float* out_w,
                      float* xz, float* xc, float* dbl, float* dtb, float* yb,
                      float* hseg, float* hin, float* sumdt, float* wpad,
                      float* outbuf, hipStream_t stream) {
    const int nchunks = rows / L;
    const int nseg    = L / SL;
    // xz = x @ in_w^T   (M x 512, WMMA)
    launch_gemm(xin, D_MODEL, in_w, D_MODEL, xz, 2 * D_INNER, nullptr,
                rows, 2 * D_INNER, D_MODEL, stream);
    // xc = silu(conv(xm))
    {
        int n = rows * D_INNER;
        conv_silu_kernel<<<(n + 255) / 256, 256, 0, stream>>>(
            xz, 2 * D_INNER, conv_w, conv_b, xc, L, rows);
    }
    // dbl = xc @ xproj_w^T  via zero-padded N=48 (WMMA)
    {
        int n = NPROJ_PAD * D_INNER;
        pad_w_kernel<<<(n + 255) / 256, 256, 0, stream>>>(xproj_w, wpad);
    }
    launch_gemm(xc, D_INNER, wpad, D_INNER, dbl, NPROJ_PAD, nullptr,
                rows, NPROJ_PAD, D_INNER, stream);
    // dt_lin = dbl[:, :8] @ dt_w^T   (M x 256, K=8, WMMA)
    launch_gemm(dbl, NPROJ_PAD, dt_w, DT_RANK, dtb, D_INNER, nullptr,
                rows, D_INNER, DT_RANK, stream);
    // dt = softplus(dt_lin + b)
    {
        int n = rows * D_INNER;
        softplus_bias_kernel<<<(n + 255) / 256, 256, 0, stream>>>(dtb, dt_b, rows);
    }
    // segmented scan (parallel over chunks x segments, 2x recompute)
    scan_pass1_kernel<<<nchunks * nseg, D_INNER, 0, stream>>>(
        dtb, xc, dbl, A_log, hseg, sumdt, L, SL, nseg);
    scan_combine_kernel<<<nchunks, D_INNER, 0, stream>>>(hseg, sumdt, A_log, hin, nseg);
    scan_pass2_kernel<<<nchunks * nseg, D_INNER, 0, stream>>>(
        dtb, xc, dbl, A_log, Dp, xz, 2 * D_INNER, hin, yb, L, SL, nseg);
    // out = y @ out_w^T   (M x 128, K=256, WMMA)
    launch_gemm(yb, D_INNER, out_w, D_INNER, outbuf, D_MODEL, nullptr,
                rows, D_MODEL, D_INNER, stream);
}

extern "C" void kernel_launch(void* const* d_in, const int* in_sizes, int n_in,
                              void* d_out, int out_size, void* d_ws, size_t ws_size,
                              hipStream_t stream) {
    const float* x         = (const float*)d_in[0];
    const float* l_in_w    = (const float*)d_in[1];
    const float* l_conv_w  = (const float*)d_in[2];
    const float* l_conv_b  = (const float*)d_in[3];
    const float* l_xproj_w = (const float*)d_in[4];
    const float* l_dt_w    = (const float*)d_in[5];
    const float* l_dt_b    = (const float*)d_in[6];
    const float* l_A_log   = (const float*)d_in[7];
    const float* l_D       = (const float*)d_in[8];
    const float* l_out_w   = (const float*)d_in[9];
    const float* g_in_w    = (const float*)d_in[10];
    const float* g_conv_w  = (const float*)d_in[11];
    const float* g_conv_b  = (const float*)d_in[12];
    const float* g_xproj_w = (const float*)d_in[13];
    const float* g_dt_w    = (const float*)d_in[14];
    const float* g_dt_b    = (const float*)d_in[15];
    const float* g_A_log   = (const float*)d_in[16];
    const float* g_D       = (const float*)d_in[17];
    const float* g_out_w   = (const float*)d_in[18];
    const float* wq        = (const float*)d_in[19];
    const float* wk        = (const float*)d_in[20];
    const float* wv        = (const float*)d_in[21];
    const float* wo        = (const float*)d_in[22];
    // d_in[23] = chunk_size (2048, baked in)

    const size_t R1 = (size_t)BATCH * SEQ;    // 16384 local rows
    const size_t R2 = (size_t)BATCH * 512;    // 2048 global rows
    const int L1 = 2048, L2 = 512, SL = 128;

    float* ws = (float*)d_ws;
    size_t off = 0;
    float* local_buf = ws + off; off += R1 * D_MODEL;
    float* gout_buf  = ws + off; off += R2 * D_MODEL;
    float* gctx      = ws + off; off += R2 * D_MODEL;
    float* wpad      = ws + off; off += (size_t)NPROJ_PAD * D_INNER;
    float* hseg      = ws + off; off += 8 * 16 * D_INNER * D_STATE;   // max chunks*segs
    float* hin       = ws + off; off += 8 * 16 * D_INNER * D_STATE;
    float* sumdt     = ws + off; off += 8 * 16 * D_INNER;
    float* xz        = ws + off; off += R1 * 2 * D_INNER;
    float* xc        = ws + off; off += R1 * D_INNER;
    float* dbl       = ws + off; off += R1 * NPROJ_PAD;
    float* dtb       = ws + off; off += R1 * D_INNER;
    float* yb        = ws + off; off += R1 * D_INNER;

    // ---- local Mamba over 2048-length chunks (8 chunks x 2048 x 128) ----
    run_mamba(x, (int)R1, L1, SL,
              l_in_w, l_conv_w, l_conv_b, l_xproj_w, l_dt_w, l_dt_b,
              l_A_log, l_D, l_out_w,
              xz, xc, dbl, dtb, yb, hseg, hin, sumdt, wpad,
              local_buf, stream);

    // ---- global Mamba over strided context (4 x 512 x 128) ----
    {
        int n = BATCH * 512 * D_MODEL;
        gather8_kernel<<<(n + 255) / 256, 256, 0, stream>>>(x, gctx);
    }
    run_mamba(gctx, (int)R2, L2, SL,
              g_in_w, g_conv_w, g_conv_b, g_xproj_w, g_dt_w, g_dt_b,
              g_A_log, g_D, g_out_w,
              xz, xc, dbl, dtb, yb, hseg, hin, sumdt, wpad,
              gout_buf, stream);

    // ---- cross attention (q from local, k/v from gout) ----
    // overlay q/o/k/v on the now-dead xz scratch region
    float* qbuf = xz;
    float* obuf = xz + R1 * D_MODEL;
    float* kbuf = xz + R1 * 2 * (size_t)D_MODEL;
    float* vbuf = kbuf + R2 * D_MODEL;

    launch_gemm(local_buf, D_MODEL, wq, D_MODEL, qbuf, D_MODEL, nullptr,
                (int)R1, D_MODEL, D_MODEL, stream);
    launch_gemm(gout_buf, D_MODEL, wk, D_MODEL, kbuf, D_MODEL, nullptr,
                (int)R2, D_MODEL, D_MODEL, stream);
    launch_gemm(gout_buf, D_MODEL, wv, D_MODEL, vbuf, D_MODEL, nullptr,
                (int)R2, D_MODEL, D_MODEL, stream);
    {
        dim3 blk(32, 4);
        dim3 grd(1024 / 4, N_HEADS);          // 1024 q-tiles, 4 per block
        attn_wmma_kernel<<<grd, blk, 0, stream>>>(qbuf, kbuf, vbuf, obuf);
    }
    // out = o @ wo^T + local   (residual fused into the WMMA epilogue)
    launch_gemm(obuf, D_MODEL, wo, D_MODEL, (float*)d_out, D_MODEL, local_buf,
                (int)R1, D_MODEL, D_MODEL, stream);
}